// FiedlerClusterer_31284541784155
// MI455X (gfx1250) — compile-verified
//
#include <hip/hip_runtime.h>
#include <hip/hip_bf16.h>
#include <math.h>

#define NROW 4096
#define DCOL 128
#define EPSQ 1e-12f
#define NITER 192          // even -> final vector lands in buf0 / Q-slot 0

// red[] slots
#define SSLOT 0            // red[0], red[1]: sum(y) per parity
#define CSLOT 2            // red[2]: c = 2*max(deg)
#define MSLOT 3            // red[3]: max(deg) bits
#define QSLOT 4            // red[4], red[5]: ||y||^2 per parity
#define DSLOT 6            // red[6]: v . (M v)

typedef __attribute__((ext_vector_type(2))) float v2f;
typedef __attribute__((ext_vector_type(8))) float v8f;

// ---------------- reduction helpers (wave32) ----------------
__device__ __forceinline__ float waveReduceSum(float s) {
#pragma unroll
  for (int off = 16; off > 0; off >>= 1) s += __shfl_xor(s, off, 32);
  return s;
}

// 256-thread block reduction; result valid in all threads; reusable.
__device__ __forceinline__ float blockReduce256(float s) {
  __shared__ float sm[256];
  sm[threadIdx.x] = s;
  __syncthreads();
#pragma unroll
  for (int off = 128; off > 0; off >>= 1) {
    if (threadIdx.x < (unsigned)off) sm[threadIdx.x] += sm[threadIdx.x + off];
    __syncthreads();
  }
  float r = sm[0];
  __syncthreads();
  return r;
}

// ---------------- setup kernels ----------------
__global__ void init_kernel(float* v, float* red) {
  int i = blockIdx.x * 256 + threadIdx.x;
  unsigned h = (unsigned)i * 2654435761u;
  h ^= h >> 16; h *= 2246822519u; h ^= h >> 13;
  v[i] = (float)(int)(h & 0xFFFFu) - 32767.5f;
  if (i < 8) red[i] = 0.0f;
}

// sq[i] = sum_k x[i,k]^2 ; one wave per row (each lane: one float4).
__global__ void rownorm_kernel(const float* __restrict__ x, float* __restrict__ sq) {
  int row  = blockIdx.x * 8 + (threadIdx.x >> 5);
  int lane = threadIdx.x & 31;
  const float4 w = ((const float4*)(x + (size_t)row * DCOL))[lane];
  float s = w.x * w.x + w.y * w.y + w.z * w.z + w.w * w.w;
  s = waveReduceSum(s);
  if (lane == 0) sq[row] = s;
}

// ---------------- WMMA Gram + distance kernel ----------------
// One 16x16 tile of W per wave; K=128 via 32 x V_WMMA_F32_16X16X4_F32.
__global__ void gram_dist_kernel(const float* __restrict__ x,
                                 const float* __restrict__ sq,
                                 float* __restrict__ W) {
  const int wave = threadIdx.x >> 5;
  const int lane = threadIdx.x & 31;
  const int tile = blockIdx.x * 8 + wave;       // 256*256 tiles
  const int tm = (tile >> 8) << 4;
  const int tn = (tile & 255) << 4;
  const int r  = lane & 15;
  const int kh = (lane >> 4) << 1;              // K sub-offset 0 / 2

  const float* xa = x + (size_t)(tm + r) * DCOL + kh;
  const float* xb = x + (size_t)(tn + r) * DCOL + kh;

  v8f c = {};
#pragma unroll
  for (int kk = 0; kk < DCOL; kk += 4) {
    v2f a, b;
    a.x = xa[kk]; a.y = xa[kk + 1];
    b.x = xb[kk]; b.y = xb[kk + 1];
    c = __builtin_amdgcn_wmma_f32_16x16x4_f32(false, a, false, b,
                                              (short)0, c, false, false);
  }

  const int col   = tn + (lane & 15);
  const int rbase = tm + ((lane >> 4) << 3);
  const float sqc = sq[col];
#pragma unroll
  for (int e = 0; e < 8; ++e) {
    const int row = rbase + e;
    float d2 = sq[row] + sqc - 2.0f * c[e];
    W[(size_t)row * NROW + col] = sqrtf(fmaxf(d2, 0.0f) + EPSQ);
  }
}

// ---------------- degree + Gershgorin bound ----------------
__global__ void deg_kernel(const float* __restrict__ W, float* __restrict__ deg,
                           unsigned* __restrict__ maxbits) {
  const int row = blockIdx.x;
  const float4* wr = (const float4*)(W + (size_t)row * NROW);
  float s = 0.f;
#pragma unroll 4
  for (int j = threadIdx.x; j < NROW / 4; j += 256) {
    float4 w = wr[j];
    s += (w.x + w.y) + (w.z + w.w);
  }
  s = blockReduce256(s);
  if (threadIdx.x == 0) {
    deg[row] = s;
    atomicMax(maxbits, __float_as_uint(s));   // deg > 0: bit order == value order
  }
}

__global__ void prep_kernel(float* red) {
  red[CSLOT] = 2.0f * __uint_as_float(__float_as_uint(red[MSLOT]));
}

// ||y0||^2 -> Q slot 0 (consumed by iteration 0)
__global__ void norm0_kernel(const float* __restrict__ y, float* red) {
  const int i = blockIdx.x * 256 + threadIdx.x;
  float s = blockReduce256(y[i] * y[i]);
  if (threadIdx.x == 0) atomicAdd(&red[QSLOT + 0], s);
}

// ---------------- power iteration on M = cI - L ----------------
// vout = M (vin / ||vin||); accumulate sum(vout) into S[p]; zero Q[1-p].
__global__ void matvec_iter_kernel(const float* __restrict__ W, const float* __restrict__ deg,
                                   float* red, const float* __restrict__ vin,
                                   float* __restrict__ vout, int p) {
  const int row = blockIdx.x;
  if (row == 0 && threadIdx.x == 0) red[QSLOT + (1 - p)] = 0.0f;
  const float inv = rsqrtf(fmaxf(red[QSLOT + p], 1e-30f));
  const float4* wr = (const float4*)(W + (size_t)row * NROW);
  const float4* v4 = (const float4*)vin;
  float s = 0.f;
#pragma unroll 4
  for (int j = threadIdx.x; j < NROW / 4; j += 256) {
    float4 w = wr[j], vv = v4[j];
    s = fmaf(w.x, vv.x, s); s = fmaf(w.y, vv.y, s);
    s = fmaf(w.z, vv.z, s); s = fmaf(w.w, vv.w, s);
  }
  s = blockReduce256(s);
  if (threadIdx.x == 0) {
    const float c = red[CSLOT];
    const float yi = inv * ((c - deg[row]) * vin[row] + s);
    vout[row] = yi;
    atomicAdd(&red[SSLOT + p], yi);
  }
}

// z -= mean (deflate ones); accumulate ||z||^2 into Q[1-p]; zero S[1-p].
__global__ void center_iter_kernel(float* __restrict__ z, float* red, int p) {
  const int i = blockIdx.x * 256 + threadIdx.x;
  if (i == 0) red[SSLOT + (1 - p)] = 0.0f;
  const float zi = z[i] - red[SSLOT + p] * (1.0f / (float)NROW);
  z[i] = zi;
  float ss = blockReduce256(zi * zi);
  if (threadIdx.x == 0) atomicAdd(&red[QSLOT + (1 - p)], ss);
}

__global__ void normalize_kernel(const float* __restrict__ z, const float* red,
                                 float* __restrict__ v, int q) {
  const int i = blockIdx.x * 256 + threadIdx.x;
  v[i] = z[i] * rsqrtf(fmaxf(red[QSLOT + q], 1e-30f));
}

// red[DSLOT] += v . (M v)   (v normalized); lambda2 = c - red[DSLOT]
__global__ void rayleigh_kernel(const float* __restrict__ W, const float* __restrict__ deg,
                                float* red, const float* __restrict__ v) {
  const int row = blockIdx.x;
  const float4* wr = (const float4*)(W + (size_t)row * NROW);
  const float4* v4 = (const float4*)v;
  float s = 0.f;
#pragma unroll 4
  for (int j = threadIdx.x; j < NROW / 4; j += 256) {
    float4 w = wr[j], vv = v4[j];
    s = fmaf(w.x, vv.x, s); s = fmaf(w.y, vv.y, s);
    s = fmaf(w.z, vv.z, s); s = fmaf(w.w, vv.w, s);
  }
  s = blockReduce256(s);
  if (threadIdx.x == 0) {
    const float mv = (red[CSLOT] - deg[row]) * v[row] + s;
    atomicAdd(&red[DSLOT], v[row] * mv);
  }
}

// ---------------- outputs ----------------
#define O_VEC 1
#define O_CLU (1 + NROW)
#define O_PX  (1 + NROW + 2 * NROW)
#define O_PA  (1 + NROW + 2 * NROW + 2 * DCOL)

__global__ void zero_pooledA_kernel(float* out) {
  if (threadIdx.x < 4) out[O_PA + threadIdx.x] = 0.0f;
}

__global__ void finalize_kernel(const float* red, float* out) {
  out[0] = red[CSLOT] - red[DSLOT];
}

__global__ void output_vec_kernel(const float* __restrict__ v, float* __restrict__ out) {
  const int i = blockIdx.x * 256 + threadIdx.x;
  const float vi = v[i];
  out[O_VEC + i] = vi;
  out[O_CLU + 2 * i + 0] = (vi < 0.0f) ? 1.0f : 0.0f;
  out[O_CLU + 2 * i + 1] = (vi > 0.0f) ? 1.0f : 0.0f;
}

// pooled_x[k,d] = sum_i C[i,k] * x[i,d] ; one block per (k,d)
__global__ void pooled_x_kernel(const float* __restrict__ x, const float* __restrict__ v,
                                float* __restrict__ out) {
  const int k = blockIdx.x >> 7;
  const int d = blockIdx.x & 127;
  float s = 0.f;
  for (int i = threadIdx.x; i < NROW; i += 256) {
    const float vi = v[i];
    const bool sel = (k == 0) ? (vi < 0.0f) : (vi > 0.0f);
    if (sel) s += x[(size_t)i * DCOL + d];
  }
  s = blockReduce256(s);
  if (threadIdx.x == 0) out[O_PX + k * DCOL + d] = s;
}

// pooled_A[k,l] = sum_ij C[i,k] A[i,j] C[j,l] ; one block per row of A
__global__ void pooled_A_kernel(const float* __restrict__ A, const float* __restrict__ v,
                                float* __restrict__ out) {
  const int i = blockIdx.x;
  const float vi = v[i];
  const float4* ar = (const float4*)(A + (size_t)i * NROW);
  const float4* v4 = (const float4*)v;
  float accN = 0.f, accP = 0.f;
#pragma unroll 4
  for (int j = threadIdx.x; j < NROW / 4; j += 256) {
    const float4 a = ar[j];
    const float4 vv = v4[j];
    accN += (vv.x < 0.f ? a.x : 0.f) + (vv.y < 0.f ? a.y : 0.f) +
            (vv.z < 0.f ? a.z : 0.f) + (vv.w < 0.f ? a.w : 0.f);
    accP += (vv.x > 0.f ? a.x : 0.f) + (vv.y > 0.f ? a.y : 0.f) +
            (vv.z > 0.f ? a.z : 0.f) + (vv.w > 0.f ? a.w : 0.f);
  }
  accN = blockReduce256(accN);
  accP = blockReduce256(accP);
  if (threadIdx.x == 0) {
    if (vi < 0.0f) {
      atomicAdd(&out[O_PA + 0], accN);
      atomicAdd(&out[O_PA + 1], accP);
    } else if (vi > 0.0f) {
      atomicAdd(&out[O_PA + 2], accN);
      atomicAdd(&out[O_PA + 3], accP);
    }
  }
}

// ---------------- launch ----------------
extern "C" void kernel_launch(void* const* d_in, const int* in_sizes, int n_in,
                              void* d_out, int out_size, void* d_ws, size_t ws_size,
                              hipStream_t stream) {
  const float* x = (const float*)d_in[0];
  const float* A = (const float*)d_in[1];
  float* out = (float*)d_out;
  float* ws  = (float*)d_ws;

  // workspace (floats): W[N*N] | sq[N] | deg[N] | buf0[N] | buf1[N] | v[N] | red[8]
  float* W    = ws;
  float* sq   = ws + (size_t)NROW * NROW;
  float* deg  = sq + NROW;
  float* buf0 = deg + NROW;
  float* buf1 = buf0 + NROW;
  float* v    = buf1 + NROW;
  float* red  = v + NROW;
  float* buf[2] = {buf0, buf1};

  init_kernel<<<NROW / 256, 256, 0, stream>>>(buf0, red);
  rownorm_kernel<<<NROW / 8, 256, 0, stream>>>(x, sq);
  gram_dist_kernel<<<(256 * 256) / 8, 256, 0, stream>>>(x, sq, W);
  deg_kernel<<<NROW, 256, 0, stream>>>(W, deg, (unsigned*)(red + MSLOT));
  prep_kernel<<<1, 1, 0, stream>>>(red);
  norm0_kernel<<<NROW / 256, 256, 0, stream>>>(buf0, red);

  for (int it = 0; it < NITER; ++it) {
    const int p = it & 1;
    matvec_iter_kernel<<<NROW, 256, 0, stream>>>(W, deg, red, buf[p], buf[1 - p], p);
    center_iter_kernel<<<NROW / 256, 256, 0, stream>>>(buf[1 - p], red, p);
  }

  const int q = NITER & 1;   // final (centered, unnormalized) vector + its Q slot
  normalize_kernel<<<NROW / 256, 256, 0, stream>>>(buf[q], red, v, q);
  rayleigh_kernel<<<NROW, 256, 0, stream>>>(W, deg, red, v);

  zero_pooledA_kernel<<<1, 32, 0, stream>>>(out);
  finalize_kernel<<<1, 1, 0, stream>>>(red, out);
  output_vec_kernel<<<NROW / 256, 256, 0, stream>>>(v, out);
  pooled_x_kernel<<<256, 256, 0, stream>>>(x, v, out);
  pooled_A_kernel<<<NROW, 256, 0, stream>>>(A, v, out);
}